// WaveletTransform_26104811225175
// MI455X (gfx1250) — compile-verified
//
#include <hip/hip_runtime.h>

// out = 0.5f * (3a - b - c - d) over 2x2 blocks of each (B,C) image plane.
// Input  (32,3,1024,1024) f32  -> 402.7 MB read  (streamed once -> TH_LOAD_NT)
// Output (32,3, 512, 512) f32  -> 100.7 MB write (write-once    -> TH_STORE_NT)
// Memory-bound: ~503 MB / 23.3 TB/s ~= 21.6 us floor on MI455X.

typedef __attribute__((ext_vector_type(2))) float vf2;
typedef __attribute__((ext_vector_type(4))) float vf4;

__global__ __launch_bounds__(256) void haar_ll_async_kernel(
    const float* __restrict__ x, float* __restrict__ out) {
  // Two full input rows of 1024 floats staged in LDS (8 KB).
  __shared__ __align__(16) float tile[2048];

  const unsigned t = threadIdx.x;   // 0..255
  const unsigned r = blockIdx.x;    // global output-row id, 0..49151

  // Map output row -> (image plane, row-in-plane). 512 output rows per plane.
  const unsigned image = r >> 9;
  const unsigned orow  = r & 511u;
  const size_t inRowFlt = (size_t)(image * 1024u + orow * 2u) * 1024u;

  // Uniform 64-bit base for GVS-mode async loads (SGPR pair),
  // per-lane 32-bit byte offset in a VGPR.
  const char* gbase = (const char*)(x + inRowFlt);
  const unsigned ldsbase = (unsigned)(size_t)(void*)tile; // low 32 bits = LDS addr
  const unsigned lds0 = ldsbase + t * 16u;          // row 2*orow
  const unsigned lds1 = ldsbase + 4096u + t * 16u;  // row 2*orow+1
  const unsigned voff = t * 16u;

  // CDNA5 async path: stream both rows global->LDS, bypassing VGPRs.
  // NT hint: input is read exactly once (402 MB > 192 MB L2), don't cache it.
  // Each lane writes exactly the LDS bytes it later reads, so a per-wave
  // s_wait_asynccnt is sufficient (no workgroup barrier).
  asm volatile(
      "global_load_async_to_lds_b128 %0, %2, %3 th:TH_LOAD_NT\n\t"
      "global_load_async_to_lds_b128 %1, %2, %3 offset:4096 th:TH_LOAD_NT\n\t"
      "s_wait_asynccnt 0"
      :
      : "v"(lds0), "v"(lds1), "v"(voff), "s"(gbase)
      : "memory");

  // Row 0 chunk: (a0, b0, a1, b1); Row 1 chunk: (c0, d0, c1, d1)
  const vf4 r0 = *(const vf4*)(tile + t * 4u);
  const vf4 r1 = *(const vf4*)(tile + 1024u + t * 4u);

  vf2 o;
  o.x = 0.5f * (3.0f * r0.x - r0.y - r1.x - r1.y);
  o.y = 0.5f * (3.0f * r0.z - r0.w - r1.z - r1.w);

  // Write-once output: non-temporal so it doesn't evict streamed input lines.
  __builtin_nontemporal_store(o, (vf2*)(out + (size_t)r * 512u + t * 2u));
}

extern "C" void kernel_launch(void* const* d_in, const int* in_sizes, int n_in,
                              void* d_out, int out_size, void* d_ws, size_t ws_size,
                              hipStream_t stream) {
  const float* x = (const float*)d_in[0];
  float* out = (float*)d_out;

  // rows_in = N / W; here W = 1024 per the reference.
  const long long n = (long long)in_sizes[0];
  const int rows_in = (int)(n / 1024);   // 98304
  const int rows_out = rows_in / 2;      // 49152 output rows of 512 floats

  haar_ll_async_kernel<<<dim3((unsigned)rows_out), dim3(256), 0, stream>>>(x, out);
}